// QuantumAttention_65481071406286
// MI455X (gfx1250) — compile-verified
//
#include <hip/hip_runtime.h>
#include <cstdint>

// out[i] = cos(x[i]) * cos(params[(i % 1024)*3])   -- pure streaming elementwise.
// 512 MiB of HBM traffic @ 23.3 TB/s => ~22.5 us floor; bandwidth-bound, no matmul,
// so the CDNA5 path exercised is the async global->LDS data mover (ASYNCcnt) with a
// depth-4 per-wave pipeline, NT load/store temporal hints, and v_cos_f32 TRANS math.

typedef float v4f __attribute__((ext_vector_type(4)));

#define EMBED          1024
#define COLS4          256      // float4 columns per row == threads per block
#define DEPTH          4        // async pipeline depth (LDS stages)
#define ROWS_PER_BLOCK 16

// ---- prepass: scale[e] = cos(params[e*3]) for e in [0,1024) ----------------
__global__ void qa_scale_kernel(const float* __restrict__ params,
                                float* __restrict__ scale, int n) {
    int i = blockIdx.x * blockDim.x + threadIdx.x;
    if (i < n) scale[i] = __cosf(params[3 * i]);
}

__device__ __forceinline__ void async_ld_b128_nt(uint32_t lds_off, uint64_t gaddr) {
    // Input is streamed exactly once: NT hint keeps the 256 MiB stream from
    // evicting the 192 MB L2's useful contents.
    asm volatile("global_load_async_to_lds_b128 %0, %1, off th:TH_LOAD_NT"
                 :: "v"(lds_off), "v"(gaddr) : "memory");
}

// ---- main streaming kernel --------------------------------------------------
__global__ __launch_bounds__(COLS4)
void qa_main_kernel(const float* __restrict__ x,
                    const float* __restrict__ scale,
                    float* __restrict__ out, int rows) {
    __shared__ v4f stage[DEPTH][COLS4];   // 16 KB: per-thread private slots, no barriers
    const int tid = threadIdx.x;

    // Loop-invariant per-column scale (4 KB table, L2-resident after first block).
    const v4f s4 = *reinterpret_cast<const v4f*>(scale + 4 * tid);

    // Rows for this block: r = blockIdx.x + k*gridDim.x, k = 0..cnt-1 (uniform per block).
    const int cnt = (rows - (int)blockIdx.x + (int)gridDim.x - 1) / (int)gridDim.x;
    if (cnt <= 0) return;

    const uint64_t row_stride = (uint64_t)gridDim.x * (EMBED * sizeof(float));
    const uint64_t lane_byte  = (uint64_t)(4 * tid) * sizeof(float);

    uint64_t ga_issue = (uint64_t)(uintptr_t)x
                      + (uint64_t)blockIdx.x * (EMBED * sizeof(float)) + lane_byte;
    uint64_t ga_store = (uint64_t)(uintptr_t)out
                      + (uint64_t)blockIdx.x * (EMBED * sizeof(float)) + lane_byte;

    const uint32_t lds_base = (uint32_t)(uintptr_t)&stage[0][tid]; // low 32b of flat = LDS addr

    // ---- prologue: fill the async pipeline ----
    const int pre = cnt < DEPTH ? cnt : DEPTH;
    for (int s = 0; s < pre; ++s) {
        async_ld_b128_nt(lds_base + (uint32_t)s * (COLS4 * sizeof(v4f)), ga_issue);
        ga_issue += row_stride;
    }

    // ---- steady state: pipeline full, constant wait immediate ----
    const int steady = cnt - DEPTH;            // may be negative if cnt < DEPTH
    int k = 0;
#pragma unroll 4
    for (; k < steady; ++k) {
        const uint32_t slot_off = lds_base + (uint32_t)(k & (DEPTH - 1)) * (COLS4 * sizeof(v4f));
        // Async loads complete in order per wave: asynccnt<=3 => oldest (load k) landed.
        asm volatile("s_wait_asynccnt 3" ::: "memory");
        const v4f xv = stage[k & (DEPTH - 1)][tid];          // ds_load_b128 (own slot only)
        v4f o;
        o.x = __cosf(xv.x) * s4.x;                           // v_cos_f32 (TRANS class)
        o.y = __cosf(xv.y) * s4.y;
        o.z = __cosf(xv.z) * s4.z;
        o.w = __cosf(xv.w) * s4.w;
        __builtin_nontemporal_store(o, reinterpret_cast<v4f*>((uintptr_t)ga_store));
        ga_store += row_stride;
        // DS and ASYNC queues are mutually unordered: retire the ds_load of this slot
        // before letting the async engine overwrite it (WAR on LDS).
        asm volatile("s_wait_dscnt 0" ::: "memory");
        async_ld_b128_nt(slot_off, ga_issue);
        ga_issue += row_stride;
    }

    // ---- drain: <= DEPTH iterations, wait for everything ----
    for (; k < cnt; ++k) {
        asm volatile("s_wait_asynccnt 0" ::: "memory");
        const v4f xv = stage[k & (DEPTH - 1)][tid];
        v4f o;
        o.x = __cosf(xv.x) * s4.x;
        o.y = __cosf(xv.y) * s4.y;
        o.z = __cosf(xv.z) * s4.z;
        o.w = __cosf(xv.w) * s4.w;
        __builtin_nontemporal_store(o, reinterpret_cast<v4f*>((uintptr_t)ga_store));
        ga_store += row_stride;
    }
}

// ---- launcher ---------------------------------------------------------------
extern "C" void kernel_launch(void* const* d_in, const int* in_sizes, int n_in,
                              void* d_out, int out_size, void* d_ws, size_t ws_size,
                              hipStream_t stream) {
    const float* x      = (const float*)d_in[0];   // [8, 8192, 1024] fp32
    const float* params = (const float*)d_in[1];   // [16, 64, 3]    fp32
    float* out   = (float*)d_out;
    float* scale = (float*)d_ws;                   // 1024 floats of scratch

    const int total = in_sizes[0];
    const int rows  = total / EMBED;               // 65536

    qa_scale_kernel<<<(EMBED + 255) / 256, 256, 0, stream>>>(params, scale, EMBED);

    const int blocks = (rows + ROWS_PER_BLOCK - 1) / ROWS_PER_BLOCK;  // 4096
    qa_main_kernel<<<blocks, COLS4, 0, stream>>>(x, scale, out, rows);
}